// FairDisCo_80204219285676
// MI455X (gfx1250) — compile-verified
//
#include <hip/hip_runtime.h>
#include <hip/hip_bf16.h>
#include <math.h>

typedef __attribute__((ext_vector_type(2))) float v2f;
typedef __attribute__((ext_vector_type(8))) float v8f;
typedef __attribute__((ext_vector_type(4))) unsigned int v4u;
typedef __attribute__((ext_vector_type(8))) unsigned int v8u;

#define NROWS 4096
#define XD 128
#define HD 512
#define ZD 8

// D = A(16x4,f32) * B(4x16,f32) + C(16x16,f32), wave32, one matrix per wave.
__device__ __forceinline__ v8f wmma_k4(v2f a, v2f b, v8f c) {
    return __builtin_amdgcn_wmma_f32_16x16x4_f32(false, a, false, b, (short)0, c,
                                                 false, false);
}

// Tensor Data Mover: 1D tile of `elems` f32 from global -> LDS byte offset.
// D# per CDNA5 ISA (group0: count/lds/global/type,
// group1: data_size=4B, tensor_dim0=tile_dim0=elems, dim1=1).
__device__ __forceinline__ void tdm_load_f32(const float* gsrc, unsigned lds_off,
                                             unsigned elems) {
    unsigned long long ga = (unsigned long long)(uintptr_t)gsrc;
    v4u g0;
    g0[0] = 1u;                                  // count=1, user descriptor
    g0[1] = lds_off;                             // lds_addr (bytes)
    g0[2] = (unsigned)ga;                        // global_addr[31:0]
    g0[3] = (unsigned)(ga >> 32) | (2u << 30);   // global_addr[56:32] | type=2
    v8u g1;
    g1[0] = 2u << 16;                            // data_size = 4 bytes
    g1[1] = (elems & 0xFFFFu) << 16;             // tensor_dim0[15:0]
    g1[2] = (elems >> 16) | (1u << 16);          // tensor_dim0[31:16] | dim1=1
    g1[3] = (elems & 0xFFFFu) << 16;             // tile_dim0 = elems
    g1[4] = 1u;                                  // tile_dim1 = 1
    g1[5] = elems;                               // tensor_dim0_stride
    g1[6] = 0u;
    g1[7] = 0u;
    asm volatile("tensor_load_to_lds %0, %1" :: "s"(g0), "s"(g1) : "memory");
}

// ---------------------------------------------------------------------------
// accumulator layout in ws (floats): [0]=kl_sum [1]=re_sum [2..17]=S[4][4]
//                                    [18..21]=group counts
// ---------------------------------------------------------------------------
__global__ void k_init(float* __restrict__ acc) {
    if (threadIdx.x < 22) acc[threadIdx.x] = 0.f;
}

// h = relu(x @ W1[:128] + b1 + W1[128+s])   -> h[4096,512]
__global__ void k_enc(const float* __restrict__ x, const float* __restrict__ W1,
                      const float* __restrict__ b1, const int* __restrict__ s,
                      float* __restrict__ h) {
    const int lane  = threadIdx.x;
    const int rbase = blockIdx.x * 16;
    const int cbase = blockIdx.y * 16;
    const int mn    = lane & 15;
    const int koff  = (lane >> 4) * 2;   // A: K = koff, koff+1 ; B likewise
    v8f acc = {0.f, 0.f, 0.f, 0.f, 0.f, 0.f, 0.f, 0.f};
#pragma unroll 4
    for (int k0 = 0; k0 < XD; k0 += 4) {
        v2f a, b;
        const float* ap = x + (rbase + mn) * XD + k0 + koff;
        a.x = ap[0];
        a.y = ap[1];
        b.x = W1[(k0 + koff) * HD + cbase + mn];
        b.y = W1[(k0 + koff + 1) * HD + cbase + mn];
        if (k0 + 4 < XD)
            __builtin_prefetch(&W1[(k0 + 4 + koff) * HD + cbase + mn], 0, 1);
        acc = wmma_k4(a, b, acc);
    }
    const int col   = cbase + mn;
    const float bc  = b1[col];
    const int rhalf = (lane >> 4) << 3;  // C: lanes 16-31 hold M = r+8
#pragma unroll
    for (int r = 0; r < 8; ++r) {
        const int row = rbase + rhalf + r;
        float v = acc[r] + bc + W1[(XD + s[row]) * HD + col];
        h[row * HD + col] = fmaxf(v, 0.f);
    }
}

// hh = h @ W2 + b2   -> hh[4096,16]
__global__ void k_hh(const float* __restrict__ h, const float* __restrict__ W2,
                     const float* __restrict__ b2, float* __restrict__ hh) {
    const int lane  = threadIdx.x;
    const int rbase = blockIdx.x * 16;
    const int mn    = lane & 15;
    const int koff  = (lane >> 4) * 2;
    v8f acc = {0.f, 0.f, 0.f, 0.f, 0.f, 0.f, 0.f, 0.f};
#pragma unroll 4
    for (int k0 = 0; k0 < HD; k0 += 4) {
        v2f a, b;
        const float* ap = h + (rbase + mn) * HD + k0 + koff;
        a.x = ap[0];
        a.y = ap[1];
        b.x = W2[(k0 + koff) * 16 + mn];
        b.y = W2[(k0 + koff + 1) * 16 + mn];
        acc = wmma_k4(a, b, acc);
    }
    const float bc  = b2[mn];
    const int rhalf = (lane >> 4) << 3;
#pragma unroll
    for (int r = 0; r < 8; ++r) {
        const int row = rbase + rhalf + r;
        hh[row * 16 + mn] = acc[r] + bc;
    }
}

// mean/var/z + per-row KL + group counts
__global__ void k_latent(const float* __restrict__ hh, const float* __restrict__ noise,
                         const int* __restrict__ s, float* __restrict__ meanv,
                         float* __restrict__ varv, float* __restrict__ zv,
                         float* __restrict__ acc) {
    __shared__ float red[256];
    const int row = blockIdx.x * blockDim.x + threadIdx.x;
    float kl = 0.f;
#pragma unroll
    for (int d = 0; d < ZD; ++d) {
        float m  = hh[row * 16 + d];
        float lv = hh[row * 16 + 8 + d];
        float v  = expf(lv);
        meanv[row * ZD + d] = m;
        varv[row * ZD + d]  = v;
        zv[row * ZD + d]    = m + noise[row * ZD + d] * sqrtf(v);
        kl += -0.5f * (1.f + lv - m * m - v);
    }
    atomicAdd(&acc[18 + s[row]], 1.f);
    red[threadIdx.x] = kl;
    __syncthreads();
    for (int off = 128; off > 0; off >>= 1) {
        if (threadIdx.x < off) red[threadIdx.x] += red[threadIdx.x + off];
        __syncthreads();
    }
    if (threadIdx.x == 0) atomicAdd(&acc[0], red[0]);
}

// g = relu(z @ W3[:8] + b3 + W3[8+s])   -> g[4096,512]
__global__ void k_g(const float* __restrict__ zv, const float* __restrict__ W3,
                    const float* __restrict__ b3, const int* __restrict__ s,
                    float* __restrict__ g) {
    const int lane  = threadIdx.x;
    const int rbase = blockIdx.x * 16;
    const int cbase = blockIdx.y * 16;
    const int mn    = lane & 15;
    const int koff  = (lane >> 4) * 2;
    v8f acc = {0.f, 0.f, 0.f, 0.f, 0.f, 0.f, 0.f, 0.f};
#pragma unroll
    for (int k0 = 0; k0 < ZD; k0 += 4) {
        v2f a, b;
        const float* ap = zv + (rbase + mn) * ZD + k0 + koff;
        a.x = ap[0];
        a.y = ap[1];
        b.x = W3[(k0 + koff) * HD + cbase + mn];
        b.y = W3[(k0 + koff + 1) * HD + cbase + mn];
        acc = wmma_k4(a, b, acc);
    }
    const int col   = cbase + mn;
    const float bc  = b3[col];
    const int rhalf = (lane >> 4) << 3;
#pragma unroll
    for (int r = 0; r < 8; ++r) {
        const int row = rbase + rhalf + r;
        float v = acc[r] + bc + W3[(ZD + s[row]) * HD + col];
        g[row * HD + col] = fmaxf(v, 0.f);
    }
}

// out = g @ W4 + b4   -> out[4096,128]
__global__ void k_out(const float* __restrict__ g, const float* __restrict__ W4,
                      const float* __restrict__ b4, float* __restrict__ outl) {
    const int lane  = threadIdx.x;
    const int rbase = blockIdx.x * 16;
    const int cbase = blockIdx.y * 16;
    const int mn    = lane & 15;
    const int koff  = (lane >> 4) * 2;
    v8f acc = {0.f, 0.f, 0.f, 0.f, 0.f, 0.f, 0.f, 0.f};
#pragma unroll 4
    for (int k0 = 0; k0 < HD; k0 += 4) {
        v2f a, b;
        const float* ap = g + (rbase + mn) * HD + k0 + koff;
        a.x = ap[0];
        a.y = ap[1];
        b.x = W4[(k0 + koff) * XD + cbase + mn];
        b.y = W4[(k0 + koff + 1) * XD + cbase + mn];
        if (k0 + 4 < HD)
            __builtin_prefetch(&W4[(k0 + 4 + koff) * XD + cbase + mn], 0, 1);
        acc = wmma_k4(a, b, acc);
    }
    const int col   = cbase + mn;
    const float bc  = b4[col];
    const int rhalf = (lane >> 4) << 3;
#pragma unroll
    for (int r = 0; r < 8; ++r) {
        const int row = rbase + rhalf + r;
        outl[row * XD + col] = acc[r] + bc;
    }
}

// re_loss: per (row, group of 8) log-softmax NLL, summed
__global__ void k_re(const float* __restrict__ outl, const int* __restrict__ c,
                     float* __restrict__ acc) {
    __shared__ float red[256];
    const int idx = blockIdx.x * blockDim.x + threadIdx.x;  // 0..65535
    const int row = idx >> 4;
    const int grp = idx & 15;
    const float* lg = outl + row * XD + grp * 8;
    float mx = lg[0];
#pragma unroll
    for (int j = 1; j < 8; ++j) mx = fmaxf(mx, lg[j]);
    float se = 0.f;
#pragma unroll
    for (int j = 0; j < 8; ++j) se += expf(lg[j] - mx);
    const int ci = c[row * 16 + grp];
    const float ls = lg[ci] - mx - logf(se);
    red[threadIdx.x] = -ls;
    __syncthreads();
    for (int off = 128; off > 0; off >>= 1) {
        if (threadIdx.x < off) red[threadIdx.x] += red[threadIdx.x + off];
        __syncthreads();
    }
    if (threadIdx.x == 0) atomicAdd(&acc[1], red[0]);
}

// pairwise Gaussian kernel, accumulate S[s_i][s_j] += K(i,j) over 64x64 tiles.
// mean/var tiles staged via TDM; s tiles via async global->LDS loads.
__global__ void k_pair(const float* __restrict__ meanv, const float* __restrict__ varv,
                       const int* __restrict__ s, float* __restrict__ acc) {
    __shared__ float im[512], iv[512], jm[512], jv[512];
    __shared__ int is_[64], js_[64];
    __shared__ float Sacc[16];
    const int t     = threadIdx.x;
    const int ibase = blockIdx.x * 64;
    const int jbase = blockIdx.y * 64;

    // TDM: 4 x 2KB tile loads, issued by wave 0 (EXEC ignored by tensor ops;
    // any duplicate issue writes identical bytes and is harmless).
    if (t < 32) {
        tdm_load_f32(meanv + ibase * ZD, (unsigned)(uintptr_t)&im[0], 512u);
        tdm_load_f32(varv  + ibase * ZD, (unsigned)(uintptr_t)&iv[0], 512u);
        tdm_load_f32(meanv + jbase * ZD, (unsigned)(uintptr_t)&jm[0], 512u);
        tdm_load_f32(varv  + jbase * ZD, (unsigned)(uintptr_t)&jv[0], 512u);
    }
    // async global->LDS for the sensitive-attribute tiles
    if (t < 64) {
        const int* gi = s + ibase + t;
        const int* gj = s + jbase + t;
        unsigned li = (unsigned)(uintptr_t)&is_[t];
        unsigned lj = (unsigned)(uintptr_t)&js_[t];
        asm volatile("global_load_async_to_lds_b32 %0, %1, off"
                     :: "v"(li), "v"(gi) : "memory");
        asm volatile("global_load_async_to_lds_b32 %0, %1, off"
                     :: "v"(lj), "v"(gj) : "memory");
    }
    if (t < 16) Sacc[t] = 0.f;
    __builtin_amdgcn_s_wait_tensorcnt(0);
    asm volatile("s_wait_asynccnt 0x0" ::: "memory");
    __syncthreads();

    const int li = t & 63;
    const int jo = t >> 6;  // 0..3
    const int si = is_[li];
#pragma unroll 4
    for (int jj = 0; jj < 16; ++jj) {
        const int lj = jo + jj * 4;
        float quad = 0.f, prod = 1.f;
#pragma unroll
        for (int d = 0; d < ZD; ++d) {
            float dm = im[li * ZD + d] - jm[lj * ZD + d];
            float sv = iv[li * ZD + d] + jv[lj * ZD + d];
            quad += dm * dm / sv;
            prod *= 6.2831853071795864f * sv;
        }
        float Kv = expf(-0.5f * quad) / (sqrtf(prod) + 1e-10f);
        atomicAdd(&Sacc[si * 4 + js_[lj]], Kv);
    }
    __syncthreads();
    if (t < 16) atomicAdd(&acc[2 + t], Sacc[t]);
}

// combine the scalar pieces
__global__ void k_final(const float* __restrict__ acc, float* __restrict__ out) {
    const float n  = (float)NROWS;
    const float kl = acc[0] / n;
    const float re = acc[1] / n;
    float total = 0.f;
    for (int a = 0; a < 16; ++a) total += acc[2 + a];
    const float item1 = total / (n * n);
    float dis = 0.f;
    for (int i = 0; i < 4; ++i) {
        const float ni   = acc[18 + i];
        const float safe = fmaxf(ni, 1.f);
        const float Sii  = acc[2 + i * 4 + i];
        float Ri = 0.f;
        for (int b = 0; b < 4; ++b) Ri += acc[2 + i * 4 + b];
        const float cur = item1 + Sii / (safe * safe) - 2.f * Ri / (safe * n);
        const float w   = (ni > 0.f && ni < n) ? (ni / n) * (ni / n) : 0.f;
        dis += w * cur;
    }
    out[0] = re + kl + 2.0f * dis;
}

extern "C" void kernel_launch(void* const* d_in, const int* in_sizes, int n_in,
                              void* d_out, int out_size, void* d_ws, size_t ws_size,
                              hipStream_t stream) {
    const float* x     = (const float*)d_in[0];
    const float* noise = (const float*)d_in[1];
    const float* W1    = (const float*)d_in[2];
    const float* b1    = (const float*)d_in[3];
    const float* W2    = (const float*)d_in[4];
    const float* b2    = (const float*)d_in[5];
    const float* W3    = (const float*)d_in[6];
    const float* b3    = (const float*)d_in[7];
    const float* W4    = (const float*)d_in[8];
    const float* b4    = (const float*)d_in[9];
    const int*   c     = (const int*)d_in[10];
    const int*   s     = (const int*)d_in[11];

    float* ws    = (float*)d_ws;
    float* h     = ws;                       // 4096*512
    float* hh    = h    + NROWS * HD;        // 4096*16
    float* meanv = hh   + NROWS * 16;        // 4096*8
    float* varv  = meanv + NROWS * ZD;       // 4096*8
    float* zv    = varv + NROWS * ZD;        // 4096*8
    float* g     = zv   + NROWS * ZD;        // 4096*512
    float* outl  = g    + NROWS * HD;        // 4096*128
    float* acc   = outl + NROWS * XD;        // 22 floats

    k_init  <<<1, 32, 0, stream>>>(acc);
    k_enc   <<<dim3(NROWS / 16, HD / 16), 32, 0, stream>>>(x, W1, b1, s, h);
    k_hh    <<<NROWS / 16, 32, 0, stream>>>(h, W2, b2, hh);
    k_latent<<<NROWS / 256, 256, 0, stream>>>(hh, noise, s, meanv, varv, zv, acc);
    k_g     <<<dim3(NROWS / 16, HD / 16), 32, 0, stream>>>(zv, W3, b3, s, g);
    k_out   <<<dim3(NROWS / 16, XD / 16), 32, 0, stream>>>(g, W4, b4, outl);
    k_re    <<<(NROWS * 16) / 256, 256, 0, stream>>>(outl, c, acc);
    k_pair  <<<dim3(NROWS / 64, NROWS / 64), 256, 0, stream>>>(meanv, varv, s, acc);
    k_final <<<1, 1, 0, stream>>>(acc, (float*)d_out);
}